// ContrastiveLoss_23459111371346
// MI455X (gfx1250) — compile-verified
//
#include <hip/hip_runtime.h>
#include <hip/hip_bf16.h>

typedef float v2f __attribute__((ext_vector_type(2)));
typedef float v8f __attribute__((ext_vector_type(8)));

constexpr int   kB    = 8192;
constexpr int   kC    = 4096;
constexpr int   kD    = 1024;
constexpr float kInvT = 1.0f / 0.07f;

constexpr int BM     = 32;          // rows per workgroup (owns ALL of C -> deterministic row sums)
constexpr int BN     = 256;         // columns per n-step (8 waves, each 32x32 register tile)
constexpr int KC     = 32;          // K chunk staged in LDS
constexpr int KPAD   = KC + 4;      // stride 36 = 4*9 mod 64 -> conflict-free b64 frag loads
constexpr int NSTEPS = kC / BN;     // 16

// ---------------- Kernel 1: inverse L2 norms (one wave per row) ----------------
__global__ __launch_bounds__(256) void cl_norms(const float* __restrict__ a,
                                                const float* __restrict__ t,
                                                float* __restrict__ inv_na,
                                                float* __restrict__ inv_nt) {
  const int lane = threadIdx.x & 31;
  const int wid  = threadIdx.x >> 5;
  const int row  = blockIdx.x * 8 + wid;       // (kB + kC)/8 blocks
  const float* src;
  float* dst;
  if (row < kB) { src = a + (size_t)row * kD;                dst = inv_na + row; }
  else          { int r = row - kB; src = t + (size_t)r * kD; dst = inv_nt + r;  }
  float s = 0.0f;
  #pragma unroll
  for (int k = 0; k < kD; k += 128) {
    float4 v = *(const float4*)(src + k + lane * 4);
    s += v.x * v.x + v.y * v.y + v.z * v.z + v.w * v.w;
  }
  #pragma unroll
  for (int m = 16; m >= 1; m >>= 1) s += __shfl_xor(s, m, 32);
  if (lane == 0) *dst = 1.0f / fmaxf(sqrtf(s), 1e-12f);
}

// ------- Kernel 2: fused fp32-WMMA GEMM (2x2 reg tile/wave) + loss epilogue -------
__global__ __launch_bounds__(256) void cl_gemm_loss(
    const float* __restrict__ A, const float* __restrict__ T,
    const float* __restrict__ W, const int* __restrict__ posIdx,
    const float* __restrict__ inv_na, const float* __restrict__ inv_nt,
    float* __restrict__ lossOut) {
  __shared__ float As[BM * KPAD];        //  4,608 B
  __shared__ float Ts[BN * KPAD];        // 36,864 B
  __shared__ float redN[8][BM];          // per-wave row partials
  __shared__ float redP[8][BM];

  const int tid   = threadIdx.x;
  const int lane  = tid & 31;
  const int l15   = lane & 15;
  const int hl    = lane >> 4;           // half-wave select
  const int nwv   = tid >> 5;            // 0..7 : all waves N-parallel
  const int mbase = blockIdx.x * BM;

  // rows this lane's accumulators map to: row = mbase + t*16 + 8*hl + r
  float ina[2][8]; int pp[2][8];
  #pragma unroll
  for (int t = 0; t < 2; ++t)
    #pragma unroll
    for (int r = 0; r < 8; ++r) {
      const int row = mbase + t * 16 + 8 * hl + r;
      ina[t][r] = inv_na[row];
      pp[t][r]  = posIdx[row];
    }

  float rowNeg[2][8], rowPos[2][8];
  #pragma unroll
  for (int t = 0; t < 2; ++t)
    #pragma unroll
    for (int r = 0; r < 8; ++r) { rowNeg[t][r] = 0.0f; rowPos[t][r] = 0.0f; }

  for (int ns = 0; ns < NSTEPS; ++ns) {
    const int nbase = ns * BN;
    v8f acc00 = {0,0,0,0,0,0,0,0};  // (m-tile 0, n-tile 0)
    v8f acc01 = {0,0,0,0,0,0,0,0};  // (m-tile 0, n-tile 1)
    v8f acc10 = {0,0,0,0,0,0,0,0};  // (m-tile 1, n-tile 0)
    v8f acc11 = {0,0,0,0,0,0,0,0};  // (m-tile 1, n-tile 1)

    for (int kc = 0; kc < kD; kc += KC) {
      __syncthreads();
      {                                              // 32x32 A tile: 1 float4 / thread
        const int f = tid * 4;
        const int row = f >> 5, col = f & 31;
        float4 v = *(const float4*)(A + (size_t)(mbase + row) * kD + kc + col);
        *(float4*)(&As[row * KPAD + col]) = v;
      }
      #pragma unroll
      for (int it = 0; it < 8; ++it) {               // 256x32 T tile
        const int f = (it * 256 + tid) * 4;
        const int row = f >> 5, col = f & 31;
        float4 v = *(const float4*)(T + (size_t)(nbase + row) * kD + kc + col);
        *(float4*)(&Ts[row * KPAD + col]) = v;
      }
      __syncthreads();

      const float* aR0 = &As[(0 * 16 + l15) * KPAD + 2 * hl];
      const float* aR1 = &As[(1 * 16 + l15) * KPAD + 2 * hl];
      const float* bR0 = &Ts[((nwv * 2 + 0) * 16 + l15) * KPAD + 2 * hl];
      const float* bR1 = &Ts[((nwv * 2 + 1) * 16 + l15) * KPAD + 2 * hl];
      #pragma unroll
      for (int kk = 0; kk < KC; kk += 4) {
        v2f a0 = *(const v2f*)(aR0 + kk);
        v2f a1 = *(const v2f*)(aR1 + kk);
        v2f b0 = *(const v2f*)(bR0 + kk);
        v2f b1 = *(const v2f*)(bR1 + kk);
        acc00 = __builtin_amdgcn_wmma_f32_16x16x4_f32(false, a0, false, b0,
                                                      (short)0, acc00, false, false);
        acc01 = __builtin_amdgcn_wmma_f32_16x16x4_f32(false, a0, false, b1,
                                                      (short)0, acc01, false, false);
        acc10 = __builtin_amdgcn_wmma_f32_16x16x4_f32(false, a1, false, b0,
                                                      (short)0, acc10, false, false);
        acc11 = __builtin_amdgcn_wmma_f32_16x16x4_f32(false, a1, false, b1,
                                                      (short)0, acc11, false, false);
      }
    }

    // epilogue: scale -> exp -> weighted-neg sum / pos pick, streamed NT
    const int col0 = nbase + nwv * 32 + l15;
    const int col1 = col0 + 16;
    const float it0 = inv_nt[col0];
    const float it1 = inv_nt[col1];
    #pragma unroll
    for (int t = 0; t < 2; ++t) {
      #pragma unroll
      for (int r = 0; r < 8; ++r) {
        const int    row  = mbase + t * 16 + 8 * hl + r;
        const size_t wofs = (size_t)row * kC;
        const float w0 = __builtin_nontemporal_load(W + wofs + col0);
        const float w1 = __builtin_nontemporal_load(W + wofs + col1);
        const float d0 = t ? acc10[r] : acc00[r];
        const float d1 = t ? acc11[r] : acc01[r];
        const float lg0 = d0 * ina[t][r] * it0 * kInvT;
        const float lg1 = d1 * ina[t][r] * it1 * kInvT;
        const bool p0 = (col0 == pp[t][r]);
        const bool p1 = (col1 == pp[t][r]);
        rowPos[t][r] += p0 ? lg0 : 0.0f;
        rowPos[t][r] += p1 ? lg1 : 0.0f;
        rowNeg[t][r] += p0 ? 0.0f : __expf(lg0) * (1.0f - w0);
        rowNeg[t][r] += p1 ? 0.0f : __expf(lg1) * (1.0f - w1);
      }
    }
  }

  // fixed-order xor reduction within each 16-lane half (rows live per half)
  #pragma unroll
  for (int t = 0; t < 2; ++t) {
    #pragma unroll
    for (int r = 0; r < 8; ++r) {
      float sN = rowNeg[t][r], sP = rowPos[t][r];
      #pragma unroll
      for (int m = 8; m >= 1; m >>= 1) {
        sN += __shfl_xor(sN, m, 32);
        sP += __shfl_xor(sP, m, 32);
      }
      if (l15 == 0) {
        const int lrow = t * 16 + 8 * hl + r;
        redN[nwv][lrow] = sN;
        redP[nwv][lrow] = sP;
      }
    }
  }
  __syncthreads();
  if (tid < BM) {                        // fixed-order combine across the 8 waves
    float n = 0.0f, p = 0.0f;
    #pragma unroll
    for (int wv = 0; wv < 8; ++wv) { n += redN[wv][tid]; p += redP[wv][tid]; }
    lossOut[mbase + tid] = -p + logf(n + __expf(p));
  }
}

// ---------------- Kernel 3: deterministic mean over B rows ----------------
__global__ __launch_bounds__(256) void cl_reduce(const float* __restrict__ loss,
                                                 float* __restrict__ out) {
  __shared__ float sm[256];
  float s = 0.0f;
  for (int i = threadIdx.x; i < kB; i += 256) s += loss[i];
  sm[threadIdx.x] = s;
  __syncthreads();
  for (int k = 128; k > 0; k >>= 1) {
    if (threadIdx.x < k) sm[threadIdx.x] += sm[threadIdx.x + k];
    __syncthreads();
  }
  if (threadIdx.x == 0) out[0] = sm[0] * (1.0f / kB);
}

extern "C" void kernel_launch(void* const* d_in, const int* in_sizes, int n_in,
                              void* d_out, int out_size, void* d_ws, size_t ws_size,
                              hipStream_t stream) {
  (void)in_sizes; (void)n_in; (void)out_size; (void)ws_size;
  const float* a   = (const float*)d_in[0];
  const float* t   = (const float*)d_in[1];
  const float* w   = (const float*)d_in[2];
  const int*   pos = (const int*)d_in[3];

  float* inv_na = (float*)d_ws;          // kB floats
  float* inv_nt = inv_na + kB;           // kC floats
  float* loss   = inv_nt + kC;           // kB floats

  cl_norms<<<(kB + kC) / 8, 256, 0, stream>>>(a, t, inv_na, inv_nt);
  cl_gemm_loss<<<kB / BM, 256, 0, stream>>>(a, t, w, pos, inv_na, inv_nt, loss);
  cl_reduce<<<1, 256, 0, stream>>>(loss, (float*)d_out);
}